// InterpolateSparse2d_90933047590988
// MI455X (gfx1250) — compile-verified
//
#include <hip/hip_runtime.h>

// Sparse bilinear interpolation on MI455X (gfx1250, wave32).
// Memory-bound gather: ~400 MB effective HBM traffic -> ~17 us roofline.
// Grid: (blocks_per_batch, B). One wave32 per keypoint; lane L handles
// channels 2L, 2L+1. 8 outstanding global_load_b32 per lane, coalesced
// non-temporal b64 stores. No integer division anywhere on the hot path.

typedef __attribute__((ext_vector_type(2))) float v2f;

namespace {

constexpr int kB  = 16;    // batch
constexpr int kC  = 64;    // channels
constexpr int kHx = 240;   // feature-map height
constexpr int kWx = 320;   // feature-map width
constexpr int kWavesPerBlock = 8;
constexpr int kBlock = kWavesPerBlock * 32;

__global__ __launch_bounds__(kBlock) void bilerp_gather(
    const float* __restrict__ x,     // [B, C, Hx, Wx]
    const float* __restrict__ pos,   // [B, N, 2]
    const int*   __restrict__ Hp,    // scalar image height
    const int*   __restrict__ Wp,    // scalar image width
    float*       __restrict__ out,   // [B, N, C]
    int npts)                        // N
{
  const int b    = blockIdx.y;                                   // batch index
  const int n    = blockIdx.x * kWavesPerBlock + (threadIdx.x >> 5); // point in batch
  const int lane = threadIdx.x & 31;
  if (n >= npts) return;

  const int wid = b * npts + n;                                  // flat [B,N] index

  // Uniform scalars -> scalar cache (s_load / KMcnt path).
  const float Wimg = (float)Wp[0];
  const float Himg = (float)Hp[0];

  // pos[b][n][0..1] — wave-uniform 8-byte load.
  const v2f p = reinterpret_cast<const v2f*>(pos)[(size_t)wid];

  // Match reference: pos * (dim-1) / D, floor, clamp, +1 clamp.
  const float px = p.x * (float)(kWx - 1) / Wimg;
  const float py = p.y * (float)(kHx - 1) / Himg;

  int x0 = (int)floorf(px);
  x0 = max(0, min(x0, kWx - 1));
  int y0 = (int)floorf(py);
  y0 = max(0, min(y0, kHx - 1));
  const int x1 = min(x0 + 1, kWx - 1);
  const int y1 = min(y0 + 1, kHx - 1);

  const float x0f = (float)x0, x1f = (float)x1;
  const float y0f = (float)y0, y1f = (float)y1;
  const float wa = (x1f - px) * (y1f - py);
  const float wb = (x1f - px) * (py - y0f);
  const float wc = (px - x0f) * (y1f - py);
  const float wd = (px - x0f) * (py - y0f);

  // Lane L covers channels c0 = 2L and c0+1. 32-bit offsets (max 78.6M elems).
  const int c0 = lane << 1;
  const int plane = kHx * kWx;                       // 76800
  const float* __restrict__ base0 = x + (size_t)((b * kC + c0) * plane);
  const float* __restrict__ base1 = base0 + plane;

  const int oa = y0 * kWx + x0;   // (y0,x0)
  const int ob = y1 * kWx + x0;   // (y1,x0)
  const int oc = y0 * kWx + x1;   // (y0,x1)
  const int od = y1 * kWx + x1;   // (y1,x1)

  // 8 independent loads: issued back-to-back, single s_wait_loadcnt later.
  const float Ia0 = base0[oa];
  const float Ib0 = base0[ob];
  const float Ic0 = base0[oc];
  const float Id0 = base0[od];
  const float Ia1 = base1[oa];
  const float Ib1 = base1[ob];
  const float Ic1 = base1[oc];
  const float Id1 = base1[od];

  v2f r;
  r.x = fmaf(wd, Id0, fmaf(wc, Ic0, fmaf(wb, Ib0, wa * Ia0)));
  r.y = fmaf(wd, Id1, fmaf(wc, Ic1, fmaf(wb, Ib1, wa * Ia1)));

  // out[b][n][c0..c0+1]: wave writes one contiguous 256B span.
  // Non-temporal: output is streamed once; keep L2 for the feature map.
  v2f* __restrict__ op = reinterpret_cast<v2f*>(out) + (size_t)wid * (kC / 2) + lane;
  __builtin_nontemporal_store(r, op);
}

} // anonymous namespace

extern "C" void kernel_launch(void* const* d_in, const int* in_sizes, int n_in,
                              void* d_out, int out_size, void* d_ws, size_t ws_size,
                              hipStream_t stream) {
  const float* x   = (const float*)d_in[0];
  const float* pos = (const float*)d_in[1];
  const int*   Hp  = (const int*)d_in[2];
  const int*   Wp  = (const int*)d_in[3];
  float*       out = (float*)d_out;

  const int npts = in_sizes[1] / (kB * 2);            // N per batch (=20000)
  const int blocks_per_batch = (npts + kWavesPerBlock - 1) / kWavesPerBlock;

  hipLaunchKernelGGL(bilerp_gather, dim3(blocks_per_batch, kB), dim3(kBlock),
                     0, stream, x, pos, Hp, Wp, out, npts);
}